// GATLayer_41540923687147
// MI455X (gfx1250) — compile-verified
//
#include <hip/hip_runtime.h>
#include <hip/hip_bf16.h>
#include <math.h>

#define NEG_SLOPE 0.2f

typedef float v2f __attribute__((ext_vector_type(2)));
typedef float v8f __attribute__((ext_vector_type(8)));

__device__ __forceinline__ unsigned f2ord(float f) {
  unsigned u = __float_as_uint(f);
  return (u & 0x80000000u) ? ~u : (u | 0x80000000u);
}
__device__ __forceinline__ float ord2f(unsigned u) {
  return (u & 0x80000000u) ? __uint_as_float(u & 0x7fffffffu) : __uint_as_float(~u);
}
__device__ __forceinline__ float leaky(float z) { return z > 0.f ? z : NEG_SLOPE * z; }

// ---------------- utility kernels ----------------

__global__ void zero_k(float* p, int n) {
  int i = blockIdx.x * blockDim.x + threadIdx.x;
  if (i < n) p[i] = 0.f;
}

__global__ void edge_deg_k(const int* __restrict__ ei, const float* __restrict__ ew,
                           float* cnt, float* wsum, int E) {
  int e = blockIdx.x * blockDim.x + threadIdx.x;
  if (e >= E) return;
  int d = ei[E + e];
  atomicAdd(&cnt[d], 1.f);
  atomicAdd(&wsum[d], ew[e]);
}

__global__ void mean_k(const float* cnt, const float* wsum, float* meanw, int N) {
  int n = blockIdx.x * blockDim.x + threadIdx.x;
  if (n < N) meanw[n] = wsum[n] / fmaxf(cnt[n], 1.f);
}

// ce[0],ce[1] = per-head <We1, att_edge1>; ce[2] = <We2, att_edge2>
__global__ void ce_k(const float* We1, const float* ae1,
                     const float* We2, const float* ae2, float* ce) {
  float s0 = 0.f, s1 = 0.f, s2 = 0.f;
  for (int c = 0; c < 64; ++c) {
    s0 += We1[c] * ae1[c];
    s1 += We1[64 + c] * ae1[64 + c];
    s2 += We2[c] * ae2[c];
  }
  ce[0] = s0; ce[1] = s1; ce[2] = s2;
}

// ---------------- WMMA f32 GEMM: Y[N x Ncols] = X[N x K] * W[K x Ncols] ----------------
// One wave computes one 16x16 tile; blockDim = (32, Ncols/16); gridDim.x = full 16-row tiles.
// A 16x4 frag: lane l holds A[l&15][2*(l>>4)+{0,1}]; B 4x16 frag: lane l holds
// B[2*(l>>4)+{0,1}][l&15]; C/D 16x16: lane l VGPR r holds C[8*(l>>4)+r][l&15].
__global__ void gemm_wmma_f32(const float* __restrict__ X, const float* __restrict__ W,
                              float* __restrict__ Y, int K, int Ncols) {
  const int row0 = blockIdx.x * 16;
  const int col0 = threadIdx.y * 16;
  const int l = threadIdx.x;
  const int m = l & 15;
  const int kh = (l >> 4) * 2;
  v8f c = {0.f, 0.f, 0.f, 0.f, 0.f, 0.f, 0.f, 0.f};
  const float* xr = X + (size_t)(row0 + m) * K;
  for (int kk = 0; kk < K; kk += 4) {
    v2f a, b;
    a.x = xr[kk + kh];
    a.y = xr[kk + kh + 1];
    b.x = W[(size_t)(kk + kh) * Ncols + col0 + m];
    b.y = W[(size_t)(kk + kh + 1) * Ncols + col0 + m];
    c = __builtin_amdgcn_wmma_f32_16x16x4_f32(false, a, false, b, (short)0, c, false, false);
  }
  const int rb = (l >> 4) * 8;
#pragma unroll
  for (int r = 0; r < 8; ++r)
    Y[(size_t)(row0 + rb + r) * Ncols + col0 + m] = c[r];
}

// scalar tail for rows not covered by full 16-row tiles (not launched when N%16==0)
__global__ void gemm_tail_k(const float* __restrict__ X, const float* __restrict__ W,
                            float* __restrict__ Y, int row0, int N, int K, int Ncols) {
  int idx = blockIdx.x * blockDim.x + threadIdx.x;
  int total = (N - row0) * Ncols;
  if (idx >= total) return;
  int r = row0 + idx / Ncols, cc = idx % Ncols;
  float s = 0.f;
  for (int k = 0; k < K; ++k) s += X[(size_t)r * K + k] * W[(size_t)k * Ncols + cc];
  Y[(size_t)r * Ncols + cc] = s;
}

// ---------------- attention pieces ----------------

// a_src[n,h] = <xs[n,h,:], att_s[h,:]>;  a_dst likewise. i indexes (n,h).
__global__ void attn_dots_k(const float* __restrict__ xs, const float* __restrict__ att_s,
                            const float* __restrict__ att_d, float* a_s, float* a_d,
                            int N, int H, int C) {
  int i = blockIdx.x * blockDim.x + threadIdx.x;
  if (i >= N * H) return;
  int h = i % H, n = i / H;
  const float* xp = xs + (size_t)n * H * C + h * C;
  const float* sp = att_s + h * C;
  const float* dp = att_d + h * C;
  float ss = 0.f, sd = 0.f;
  for (int c = 0; c < C; ++c) { float v = xp[c]; ss += v * sp[c]; sd += v * dp[c]; }
  a_s[i] = ss; a_d[i] = sd;
}

// m_ord[n,h] initialized with the self-loop logit (ordered-uint encoding)
__global__ void init_selfmax_k(const float* a_s, const float* a_d, const float* meanw,
                               const float* ce, int ceoff, unsigned* m_ord, int N, int H) {
  int i = blockIdx.x * blockDim.x + threadIdx.x;
  if (i >= N * H) return;
  int h = i % H, n = i / H;
  float z = leaky(a_s[i] + a_d[i] + meanw[n] * ce[ceoff + h]);
  m_ord[i] = f2ord(z);
}

// per-edge leaky logit -> store; atomicMax into dst's m_ord
__global__ void edge_logit_max_k(const int* __restrict__ ei, const float* __restrict__ ew,
                                 const float* __restrict__ a_s, const float* __restrict__ a_d,
                                 const float* ce, int ceoff, float* logit, unsigned* m_ord,
                                 int E, int H) {
  int e = blockIdx.x * blockDim.x + threadIdx.x;
  if (e >= E) return;
  int s = ei[e], d = ei[E + e];
  float w = ew[e];
  for (int h = 0; h < H; ++h) {
    float z = leaky(a_s[s * H + h] + a_d[d * H + h] + w * ce[ceoff + h]);
    logit[(size_t)e * H + h] = z;
    atomicMax(&m_ord[d * H + h], f2ord(z));
  }
}

// init denom with self-loop exp; init acc with e_self * xs[n]. i indexes (n,h,c).
__global__ void node_self_init_k(const float* a_s, const float* a_d, const float* meanw,
                                 const float* ce, int ceoff, const unsigned* m_ord,
                                 const float* __restrict__ xs, float* denom, float* acc,
                                 int N, int H, int C) {
  int i = blockIdx.x * blockDim.x + threadIdx.x;
  if (i >= N * H * C) return;
  int c = i % C, nh = i / C, h = nh % H, n = nh / H;
  float z = leaky(a_s[nh] + a_d[nh] + meanw[n] * ce[ceoff + h]);
  float es = expf(z - ord2f(m_ord[nh]));
  if (c == 0) denom[nh] = es;
  acc[i] = es * xs[i];
}

// one wave per edge: e_h = exp(logit - m[dst]); denom += e_h; acc[dst] += e_h * xs[src]
template <int H, int C>
__global__ void edge_aggregate_k(const int* __restrict__ ei, const float* __restrict__ logit,
                                 const unsigned* __restrict__ m_ord, float* denom,
                                 const float* __restrict__ xs, float* acc, int E) {
  constexpr int HC = H * C;
  constexpr int CPL = HC / 32;  // channels per lane
  int lane = threadIdx.x & 31;
  int e = (int)((blockIdx.x * (size_t)blockDim.x + threadIdx.x) >> 5);
  if (e >= E) return;
  int s = ei[e], d = ei[E + e];
  float ex[H];
#pragma unroll
  for (int h = 0; h < H; ++h)
    ex[h] = expf(logit[(size_t)e * H + h] - ord2f(m_ord[d * H + h]));
  if (lane < H) atomicAdd(&denom[d * H + lane], ex[lane]);
  const float* xp = xs + (size_t)s * HC;
  float* ap = acc + (size_t)d * HC;
  int cbase = lane * CPL;
#pragma unroll
  for (int j = 0; j < CPL; ++j) {
    int c = cbase + j;
    atomicAdd(&ap[c], ex[c / C] * xp[c]);
  }
}

__global__ void finalize_elu_k(const float* acc, const float* denom, const float* bias,
                               float* out, int N, int H, int C) {
  int i = blockIdx.x * blockDim.x + threadIdx.x;
  if (i >= N * H * C) return;
  int nh = i / C;
  float v = acc[i] / (denom[nh] + 1e-16f) + bias[i % (H * C)];
  out[i] = v > 0.f ? v : expm1f(v);
}

__global__ void finalize_out_k(const float* acc, const float* denom, const float* bias,
                               float* out, int N, int C) {
  int i = blockIdx.x * blockDim.x + threadIdx.x;
  if (i >= N * C) return;
  out[i] = acc[i] / (denom[i / C] + 1e-16f) + bias[i % C];
}

// ---------------- launch ----------------

extern "C" void kernel_launch(void* const* d_in, const int* in_sizes, int n_in,
                              void* d_out, int out_size, void* d_ws, size_t ws_size,
                              hipStream_t stream) {
  const float* x   = (const float*)d_in[0];
  const int*   ei  = (const int*)d_in[1];
  const float* ew  = (const float*)d_in[2];
  const float* W1  = (const float*)d_in[3];
  const float* as1w = (const float*)d_in[4];
  const float* ad1w = (const float*)d_in[5];
  const float* We1 = (const float*)d_in[6];
  const float* ae1 = (const float*)d_in[7];
  const float* b1  = (const float*)d_in[8];
  const float* W2  = (const float*)d_in[9];
  const float* as2w = (const float*)d_in[10];
  const float* ad2w = (const float*)d_in[11];
  const float* We2 = (const float*)d_in[12];
  const float* ae2 = (const float*)d_in[13];
  const float* b2  = (const float*)d_in[14];
  float* out = (float*)d_out;

  const int N = in_sizes[0] / 128;     // 50000
  const int E = in_sizes[2];           // 800000

  float* ws = (float*)d_ws;
  size_t o = 0;
  float* cnt   = ws + o; o += N;           // zeroed (with wsum, contiguous)
  float* wsum  = ws + o; o += N;
  float* meanw = ws + o; o += N;
  float* ce    = ws + o; o += 4;
  float* xs1   = ws + o; o += (size_t)N * 128;  // reused as h1 in-place
  float* as1   = ws + o; o += (size_t)N * 2;
  float* ad1   = ws + o; o += (size_t)N * 2;
  unsigned* m1 = (unsigned*)(ws + o); o += (size_t)N * 2;
  float* den1  = ws + o; o += (size_t)N * 2;
  float* lg1   = ws + o; o += (size_t)E * 2;    // reused as lg2 (needs E)
  float* acc1  = ws + o; o += (size_t)N * 128;  // reused as acc2 (needs N*64)
  float* xs2   = ws + o; o += (size_t)N * 64;
  float* as2   = ws + o; o += N;
  float* ad2   = ws + o; o += N;
  unsigned* m2 = (unsigned*)(ws + o); o += N;
  float* den2  = ws + o; o += N;

  const int B = 256;
  auto cdiv = [](long long a, long long b) { return (int)((a + b - 1) / b); };

  // self-loop edge attr = mean of incoming edge weights
  zero_k<<<cdiv(2 * N, B), B, 0, stream>>>(cnt, 2 * N);
  edge_deg_k<<<cdiv(E, B), B, 0, stream>>>(ei, ew, cnt, wsum, E);
  mean_k<<<cdiv(N, B), B, 0, stream>>>(cnt, wsum, meanw, N);
  ce_k<<<1, 1, 0, stream>>>(We1, ae1, We2, ae2, ce);

  const int nrt = N / 16;
  // ---------------- layer 1 (H=2, C=64) ----------------
  gemm_wmma_f32<<<dim3(nrt), dim3(32, 8), 0, stream>>>(x, W1, xs1, 128, 128);
  if (N % 16)
    gemm_tail_k<<<cdiv((long long)(N % 16) * 128, B), B, 0, stream>>>(x, W1, xs1, nrt * 16, N, 128, 128);
  attn_dots_k<<<cdiv(2 * N, B), B, 0, stream>>>(xs1, as1w, ad1w, as1, ad1, N, 2, 64);
  init_selfmax_k<<<cdiv(2 * N, B), B, 0, stream>>>(as1, ad1, meanw, ce, 0, m1, N, 2);
  edge_logit_max_k<<<cdiv(E, B), B, 0, stream>>>(ei, ew, as1, ad1, ce, 0, lg1, m1, E, 2);
  node_self_init_k<<<cdiv((long long)N * 128, B), B, 0, stream>>>(as1, ad1, meanw, ce, 0, m1,
                                                                  xs1, den1, acc1, N, 2, 64);
  edge_aggregate_k<2, 64><<<cdiv((long long)E * 32, B), B, 0, stream>>>(ei, lg1, m1, den1,
                                                                        xs1, acc1, E);
  finalize_elu_k<<<cdiv((long long)N * 128, B), B, 0, stream>>>(acc1, den1, b1, xs1, N, 2, 64);

  // ---------------- layer 2 (H=1, C=64) ----------------
  gemm_wmma_f32<<<dim3(nrt), dim3(32, 4), 0, stream>>>(xs1, W2, xs2, 128, 64);
  if (N % 16)
    gemm_tail_k<<<cdiv((long long)(N % 16) * 64, B), B, 0, stream>>>(xs1, W2, xs2, nrt * 16, N, 128, 64);
  attn_dots_k<<<cdiv(N, B), B, 0, stream>>>(xs2, as2w, ad2w, as2, ad2, N, 1, 64);
  init_selfmax_k<<<cdiv(N, B), B, 0, stream>>>(as2, ad2, meanw, ce, 2, m2, N, 1);
  edge_logit_max_k<<<cdiv(E, B), B, 0, stream>>>(ei, ew, as2, ad2, ce, 2, lg1, m2, E, 1);
  node_self_init_k<<<cdiv((long long)N * 64, B), B, 0, stream>>>(as2, ad2, meanw, ce, 2, m2,
                                                                 xs2, den2, acc1, N, 1, 64);
  edge_aggregate_k<1, 64><<<cdiv((long long)E * 32, B), B, 0, stream>>>(ei, lg1, m2, den2,
                                                                        xs2, acc1, E);
  finalize_out_k<<<cdiv((long long)N * 64, B), B, 0, stream>>>(acc1, den2, b2, out, N, 64);
}